// WaveCell_790273982844
// MI455X (gfx1250) — compile-verified
//
#include <hip/hip_runtime.h>
#include <hip/hip_bf16.h>

// ---------------------------------------------------------------------------
// Damped 2-D wave equation step (WaveCell.forward) for MI455X / gfx1250.
//
// Memory-bound (~1.15 GB HBM traffic vs ~0.8 GFLOP => ~50us at 23.3 TB/s),
// so the kernel is built around CDNA5 data movement:
//   * GLOBAL_LOAD_ASYNC_TO_LDS_B128 (ASYNCcnt) to stage the y1 stencil tile
//   * global c/b/y2 loads issued BEFORE the async drain (latency overlap)
//   * B128 loads for c/b (L2-resident across the batch)
//   * non-temporal B128 load for y2 (stream) and NT stores for both outputs
//   * global_prefetch_b8 for the next tile row of y2
// 16-row tiles: 2 float4 outputs per thread, halo amplification 18/16.
// ---------------------------------------------------------------------------

typedef float v4f __attribute__((ext_vector_type(4)));

static constexpr int GH = 4096;           // grid height (fixed by reference)
static constexpr int GW = 4096;           // grid width  (fixed by reference)
static constexpr int GW4 = GW / 4;        // width in float4 units

static constexpr int TILE_ROWS = 16;      // rows per block (2 per thread)
static constexpr int TILE_Q    = 32;      // float4 columns per block (128 floats)
static constexpr int LDS_ROWS  = TILE_ROWS + 2;   // +1 halo row top/bottom (18)
static constexpr int LDS_Q     = TILE_Q + 2;      // +1 halo float4 left/right (34)
static constexpr int LDS_STRIDE = LDS_Q * 4;      // floats per LDS row (136)

__global__ __launch_bounds__(256) void wave_step_kernel(
    const float* __restrict__ cmap,
    const float* __restrict__ bmap,
    const float* __restrict__ y1,
    const float* __restrict__ y2,
    float* __restrict__ y_out,
    float* __restrict__ y1_copy)
{
    // Single shared array -> allocated at group-segment offset 0; async-load
    // VDST operand carries the LDS byte address relative to LDS_BASE.
    __shared__ float tile[LDS_ROWS * LDS_STRIDE];   // 9792 bytes

    const int tid   = threadIdx.x;
    const int q0    = blockIdx.x * TILE_Q;      // tile origin, float4 column
    const int row0  = blockIdx.y * TILE_ROWS;   // tile origin, row
    const int batch = blockIdx.z;

    const float* y1b = y1 + (size_t)batch * (size_t)GH * (size_t)GW;

    // ---- Stage y1 tile (+halo) into LDS with async B128 loads -------------
    for (int i = tid; i < LDS_ROWS * LDS_Q; i += 256) {
        const int lrow = i / LDS_Q;
        const int lq   = i - lrow * LDS_Q;
        const int grow = row0 + lrow - 1;       // global row   (halo: -1)
        const int gq   = q0 + lq - 1;           // global f4col (halo: -1)
        float* lp = &tile[lrow * LDS_STRIDE + lq * 4];
        if ((unsigned)grow < (unsigned)GH && (unsigned)gq < (unsigned)GW4) {
            // byte offset of this float4 inside the tile's shared array;
            // tile[] is the only LDS object => group-segment offset 0.
            const unsigned lds_off =
                (unsigned)((const char*)lp - (const char*)&tile[0]);
            const unsigned goff =
                ((unsigned)grow * (unsigned)GW4 + (unsigned)gq) * 16u;
            asm volatile("global_load_async_to_lds_b128 %0, %1, %2"
                         :
                         : "v"(lds_off), "v"(goff), "s"(y1b)
                         : "memory");
        } else {
            // zero padding outside the image (matches conv2d padding=1)
            v4f z = {0.0f, 0.0f, 0.0f, 0.0f};
            *(v4f*)lp = z;
        }
    }

    // ---- Per-thread output coordinates (two float4s: rows r and r+8) ------
    const int r  = tid >> 5;                    // 0..7   local row (first half)
    const int q  = tid & 31;                    // 0..31  local float4 col
    const int grow0 = row0 + r;
    const int grow1 = row0 + r + 8;
    const int gq    = q0 + q;

    const size_t plane = (size_t)GH * (size_t)GW;
    const size_t eidx0 = (size_t)grow0 * GW + (size_t)gq * 4;
    const size_t eidx1 = (size_t)grow1 * GW + (size_t)gq * 4;
    const size_t bidx0 = (size_t)batch * plane + eidx0;
    const size_t bidx1 = (size_t)batch * plane + eidx1;

    // Issue the independent global loads BEFORE draining ASYNCcnt: their
    // LOADcnt latency overlaps the async staging + barrier. The "memory"
    // clobbers on the asm below pin their program order here.
    const v4f cc0  = *(const v4f*)(cmap + eidx0);
    const v4f cc1  = *(const v4f*)(cmap + eidx1);
    const v4f bb0  = *(const v4f*)(bmap + eidx0);
    const v4f bb1  = *(const v4f*)(bmap + eidx1);
    const v4f y2v0 = __builtin_nontemporal_load((const v4f*)(y2 + bidx0));
    const v4f y2v1 = __builtin_nontemporal_load((const v4f*)(y2 + bidx1));

    // Speculative prefetch of the next tile-row of y2 (dropped if OOB).
    __builtin_prefetch(y2 + bidx0 + (size_t)TILE_ROWS * GW, 0, 0);

    // Each wave drains its own async transfers, then block-wide barrier.
    asm volatile("s_wait_asynccnt 0x0" ::: "memory");
    __syncthreads();

    // ---- Stencil + update for both rows -----------------------------------
    const float inv_dt2 = 1.0e6f;   // 1/dt^2,  dt = 1e-3
    const float inv_dt  = 1.0e3f;   // 1/dt
    const float inv_h2  = 0.01f;    // 1/h^2,   h = 10

    #pragma unroll
    for (int half = 0; half < 2; ++half) {
        const int lr = (half ? r + 9 : r + 1);              // LDS row of center
        const float* trow = &tile[lr * LDS_STRIDE + 4 + q * 4];
        const v4f cen = *(const v4f*)trow;
        const v4f up  = *(const v4f*)(trow - LDS_STRIDE);
        const v4f dn  = *(const v4f*)(trow + LDS_STRIDE);
        const float lf = trow[-1];
        const float rt = trow[4];

        v4f lap;
        lap.x = up.x + dn.x + lf    + cen.y - 4.0f * cen.x;
        lap.y = up.y + dn.y + cen.x + cen.z - 4.0f * cen.y;
        lap.z = up.z + dn.z + cen.y + cen.w - 4.0f * cen.z;
        lap.w = up.w + dn.w + cen.z + rt    - 4.0f * cen.w;

        const v4f cc  = half ? cc1  : cc0;
        const v4f bb  = half ? bb1  : bb0;
        const v4f y2v = half ? y2v1 : y2v0;
        const size_t bidx = half ? bidx1 : bidx0;

        const v4f bod = bb * inv_dt;                        // b/dt
        const v4f num = (2.0f * inv_dt2) * cen
                      - (inv_dt2 - bod) * y2v
                      + (cc * cc) * (lap * inv_h2);
        const v4f den = inv_dt2 + bod;
        const v4f res = num / den;

        // Outputs are never re-read -> non-temporal B128 stores.
        __builtin_nontemporal_store(res, (v4f*)(y_out + bidx));
        __builtin_nontemporal_store(cen, (v4f*)(y1_copy + bidx));
    }
}

extern "C" void kernel_launch(void* const* d_in, const int* in_sizes, int n_in,
                              void* d_out, int out_size, void* d_ws, size_t ws_size,
                              hipStream_t stream) {
    const float* cmap = (const float*)d_in[0];   // [H, W]
    const float* bmap = (const float*)d_in[1];   // [H, W]
    const float* y1   = (const float*)d_in[2];   // [B, H, W]
    const float* y2   = (const float*)d_in[3];   // [B, H, W]

    const int plane = GH * GW;                   // 16,777,216
    const int B     = in_sizes[2] / plane;       // 4

    float* y_out   = (float*)d_out;                          // output 0: y
    float* y1_copy = (float*)d_out + (size_t)B * plane;      // output 1: y1

    dim3 grid(GW / (TILE_Q * 4), GH / TILE_ROWS, B);         // (32, 256, B)
    dim3 block(256, 1, 1);
    wave_step_kernel<<<grid, block, 0, stream>>>(cmap, bmap, y1, y2,
                                                 y_out, y1_copy);
}